// WaveletLatentFeatureExtractor_14671608283488
// MI455X (gfx1250) — compile-verified
//
#include <hip/hip_runtime.h>
#include <hip/hip_bf16.h>

// 2-level 2D Haar DWT, (32,3,512,512) f32 -> (a2,h2,v2,d2 @128x128) + (h1,v1,d1 @256x256)
// Bandwidth-bound: 201 MB traffic -> ~8.6us floor at 23.3 TB/s.
// Strategy: one wave per 16x16 tile; both 1-D Haar passes as chained
// V_WMMA_F32_16X16X4_F32 (M1 = H1 * X * H1^T via Q = H1 * (H1*X)^T),
// LDS (stride-17 padded) for the inter-pass transpose and for re-gathering
// subbands into coalesced global stores.

typedef float v2f __attribute__((ext_vector_type(2)));
typedef float v8f __attribute__((ext_vector_type(8)));

#define INV_SQRT2 0.70710678118654752440f

__global__ __launch_bounds__(256) void haar2_wmma_kernel(const float* __restrict__ x,
                                                         float* __restrict__ out)
{
    const int lane = threadIdx.x & 31;
    const int wave = threadIdx.x >> 5;
    const int tile = blockIdx.x * 8 + wave;     // 0 .. 98303
    const int plane = tile >> 10;               // 0..95  (B*C planes)
    const int t     = tile & 1023;
    const int tr    = t >> 5;                   // tile row 0..31
    const int tc    = t & 31;                   // tile col 0..31

    const int mm = lane & 15;                   // row (A) / col (B,D) index
    const int hi = lane >> 4;                   // K-half selector

    __shared__ float ldsP[8][16 * 17];
    __shared__ float ldsQ[8][16 * 17];
    float* P = ldsP[wave];
    float* Q = ldsQ[wave];

    const float C = INV_SQRT2;

    // ---------------- pass 1: P(16x16) = H1 * X  (4 chained K=4 WMMAs) ----
    const float* xin = x + (size_t)plane * 262144 + (size_t)(tr * 16) * 512 + tc * 16;

    v8f acc = {};
#pragma unroll
    for (int kc = 0; kc < 4; ++kc) {
        // A = H1[:, 4kc:4kc+4]; lane holds (k0+2hi, k0+2hi+1) for row mm
        const int sel = 2 * kc + hi;
        v2f a;
        a.x = (mm == sel) ? C : ((mm == sel + 8) ?  C : 0.0f);
        a.y = (mm == sel) ? C : ((mm == sel + 8) ? -C : 0.0f);
        // B = X[4kc:4kc+4, :]; lane (col mm) loads rows r, r+1  (coalesced 64B rows)
        const int r = 4 * kc + 2 * hi;
        v2f b;
        b.x = xin[(size_t)r * 512 + mm];
        b.y = xin[(size_t)(r + 1) * 512 + mm];
        acc = __builtin_amdgcn_wmma_f32_16x16x4_f32(false, a, false, b,
                                                    (short)0, acc, false, false);
    }

    // P[m][n] row-major into LDS, stride 17 (conflict-free)
#pragma unroll
    for (int v = 0; v < 8; ++v) {
        const int m = v + 8 * hi;
        P[m * 17 + mm] = acc[v];
    }
    __syncthreads();

    // ---------------- pass 2: Q = H1 * P^T  ( = (P*H1^T)^T = M1^T ) -------
    v8f acc2 = {};
#pragma unroll
    for (int kc = 0; kc < 4; ++kc) {
        const int sel = 2 * kc + hi;
        v2f a;
        a.x = (mm == sel) ? C : ((mm == sel + 8) ?  C : 0.0f);
        a.y = (mm == sel) ? C : ((mm == sel + 8) ? -C : 0.0f);
        const int k = 4 * kc + 2 * hi;
        v2f b;                                  // Z[k][n] = P[n][k], stride-17 reads
        b.x = P[mm * 17 + k];
        b.y = P[mm * 17 + k + 1];
        acc2 = __builtin_amdgcn_wmma_f32_16x16x4_f32(false, a, false, b,
                                                     (short)0, acc2, false, false);
    }

    // Q[m][n] = M1[n][m]  -> LDS so we can re-gather subbands for coalesced stores
#pragma unroll
    for (int v = 0; v < 8; ++v) {
        const int m = v + 8 * hi;
        Q[m * 17 + mm] = acc2[v];
    }
    __syncthreads();

    // M1[i][j] = Q[j*17 + i];  blocks: aa=[i<8,j<8], h1=da=[i>=8,j<8],
    //                                  v1=ad=[i<8,j>=8], d1=dd=[i>=8,j>=8]

    // ---------------- output pointers (concatenated, reference return order)
    const size_t l2_band = (size_t)96 * 16384;   // a2/h2/v2/d2 band size
    const size_t l1_band = (size_t)96 * 65536;   // h1/v1/d1 band size
    float* a2p = out;
    float* h2p = out + l2_band;
    float* v2p = out + 2 * l2_band;
    float* d2p = out + 3 * l2_band;
    float* h1p = out + 4 * l2_band;
    float* v1p = h1p + l1_band;
    float* d1p = v1p + l1_band;

    // ---------------- level-1 detail stores: 8x8 blocks, 4 rows x 8 cols per pass
    {
        const int ci  = lane & 7;
        const int bi0 = lane >> 3;               // 0..3
        const size_t base = (size_t)plane * 65536 + (size_t)(tr * 8) * 256 + tc * 8;
#pragma unroll
        for (int pass = 0; pass < 2; ++pass) {
            const int bi = bi0 + 4 * pass;       // row in 8x8 block
            const size_t g = base + (size_t)bi * 256 + ci;
            h1p[g] = Q[ci * 17 + 8 + bi];        // M1[8+bi][ci]
            v1p[g] = Q[(8 + ci) * 17 + bi];      // M1[bi][8+ci]
            d1p[g] = Q[(8 + ci) * 17 + 8 + bi];  // M1[8+bi][8+ci]
        }
    }

    // ---------------- level-2 from aa (8x8 -> 4x4 per subband), lanes 0..15
    if (lane < 16) {
        const int p = lane >> 2;                 // output row 0..3
        const int q = lane & 3;                  // output col 0..3
        // aa[i][j] = Q[j*17 + i]
        const float aa00 = Q[(2 * q) * 17 + 2 * p];
        const float aa01 = Q[(2 * q + 1) * 17 + 2 * p];
        const float aa10 = Q[(2 * q) * 17 + 2 * p + 1];
        const float aa11 = Q[(2 * q + 1) * 17 + 2 * p + 1];
        // column (axis -1) pass then row (axis -2) pass, same op order as reference
        const float ca0 = (aa00 + aa01) * C, cd0 = (aa00 - aa01) * C;
        const float ca1 = (aa10 + aa11) * C, cd1 = (aa10 - aa11) * C;
        const size_t g = (size_t)plane * 16384 + (size_t)(tr * 4 + p) * 128 + tc * 4 + q;
        a2p[g] = (ca0 + ca1) * C;                // aa2
        h2p[g] = (ca0 - ca1) * C;                // da2 (detail along H)
        v2p[g] = (cd0 + cd1) * C;                // ad2
        d2p[g] = (cd0 - cd1) * C;                // dd2
    }
}

extern "C" void kernel_launch(void* const* d_in, const int* in_sizes, int n_in,
                              void* d_out, int out_size, void* d_ws, size_t ws_size,
                              hipStream_t stream) {
    (void)in_sizes; (void)n_in; (void)d_ws; (void)ws_size; (void)out_size;
    const float* x = (const float*)d_in[0];
    float* out = (float*)d_out;
    // 96 planes * 32*32 tiles = 98304 tiles; 8 waves (tiles) per 256-thread block
    const int blocks = 98304 / 8;                // 12288
    haar2_wmma_kernel<<<blocks, 256, 0, stream>>>(x, out);
}